// NeuralPonds_54898271977921
// MI455X (gfx1250) — compile-verified
//
#include <hip/hip_runtime.h>
#include <stdint.h>

// NeuralPonds gather kernel for MI455X (gfx1250, wave32).
//
// Memory-bound: ~192 MB of traffic / 23.3 TB/s ~= 8.3 us floor. No matmul
// structure -> no WMMA; the CDNA5 feature that fits is the async
// global<->LDS data-mover path (ASYNCcnt) for the 4 KB row gather/scatter.
//
// Per token (one wave32 each):
//   1) row-sum 1024 f32 (8 x b128 NT loads/lane-group + ds_swizzle butterfly)
//   2) flavor = int(|sum|) % 10000 ; row = pond*10000 + flavor
//   3) async-DMA the 4 KB table row: global -> LDS -> out (NT store)

#define D_MODEL          1024
#define POND_MOD         10000
#define WAVES_PER_BLOCK  8
#define BLOCK_THREADS    (WAVES_PER_BLOCK * 32)

typedef float v4f __attribute__((ext_vector_type(4)));

__global__ __launch_bounds__(BLOCK_THREADS)
void NeuralPonds_54898271977921_kernel(const float* __restrict__ ctx,     // [T, 1024]
                                       const int*   __restrict__ ponds,   // [T]
                                       const float* __restrict__ tables,  // [10, 10000, 1024]
                                       float*       __restrict__ out,     // [T, 1024]
                                       int total_tokens)
{
    // 4 KB of LDS staging per wave (32 KB per block; well under 320 KB/WGP).
    __shared__ float smem[WAVES_PER_BLOCK * D_MODEL];

    const int lane  = threadIdx.x & 31;
    const int wave  = threadIdx.x >> 5;
    const int token = blockIdx.x * WAVES_PER_BLOCK + wave;
    if (token >= total_tokens) return;   // whole wave exits together

    // ---- Phase 1: row sum of 1024 f32 (one wave per token) -------------
    // Each lane reads 8 x float4 = 32 floats; the wave reads 512 B per
    // vector load, fully coalesced. Non-temporal: context is streamed
    // exactly once -> keep L2 capacity for table rows.
    const v4f* __restrict__ crow =
        (const v4f*)(ctx + (size_t)token * D_MODEL);
    float s = 0.0f;
#pragma unroll
    for (int i = 0; i < 8; ++i) {
        v4f v = __builtin_nontemporal_load(&crow[lane + 32 * i]);
        s += (v.x + v.y) + (v.z + v.w);
    }
    // Butterfly reduction across the 32 lanes (wave32, ds_swizzle path).
#pragma unroll
    for (int m = 16; m >= 1; m >>= 1)
        s += __shfl_xor(s, m, 32);

    // int() truncation toward zero on a non-negative value == astype(int32).
    const int flavor = ((int)fabsf(s)) % POND_MOD;
    const int pond   = ponds[token];
    const size_t row = ((size_t)pond * (size_t)POND_MOD + (size_t)flavor)
                       * (size_t)D_MODEL;

    // ---- Phase 2: async-DMA the 4 KB table row: global -> LDS -> out ----
    // 32 lanes x 16 B = 512 B per issue; 8 issues cover the 1024-f32 row.
    const uint64_t src = (uint64_t)(uintptr_t)(tables + row)
                       + (uint64_t)lane * 16u;
    const uint64_t dst = (uint64_t)(uintptr_t)(out + (size_t)token * D_MODEL)
                       + (uint64_t)lane * 16u;
    // Low 32 bits of a generic pointer to __shared__ are the LDS byte offset
    // (flat LDS aperture mapping).
    const uint32_t lds0 =
        (uint32_t)(uintptr_t)(&smem[wave * D_MODEL]) + (uint32_t)lane * 16u;

#pragma unroll
    for (int i = 0; i < 8; ++i) {
        uint32_t l = lds0 + (uint32_t)i * 512u;
        uint64_t g = src  + (uint64_t)i * 512u;
        asm volatile("global_load_async_to_lds_b128 %0, %1, off"
                     :: "v"(l), "v"(g) : "memory");
    }
    // LDS RAW fence: staging data must land before the store side reads it.
    asm volatile("s_wait_asynccnt 0x0" ::: "memory");
#pragma unroll
    for (int i = 0; i < 8; ++i) {
        uint32_t l = lds0 + (uint32_t)i * 512u;
        uint64_t g = dst  + (uint64_t)i * 512u;
        // Output is written once and never re-read on device -> non-temporal,
        // don't displace table rows in L2.
        asm volatile("global_store_async_from_lds_b128 %0, %1, off th:TH_STORE_NT"
                     :: "v"(g), "v"(l) : "memory");
    }
    // No trailing wait needed: s_endpgm performs an implicit wait-idle.
}

extern "C" void kernel_launch(void* const* d_in, const int* in_sizes, int n_in,
                              void* d_out, int out_size, void* d_ws, size_t ws_size,
                              hipStream_t stream) {
    (void)n_in; (void)out_size; (void)d_ws; (void)ws_size;
    const float* ctx    = (const float*)d_in[0];  // [B,S,1024] f32
    const int*   ponds  = (const int*)  d_in[1];  // [B,S] i32
    const float* tables = (const float*)d_in[2];  // [10,10000,1024] f32
    float*       out    = (float*)d_out;          // [B,S,1024] f32

    const int total_tokens = in_sizes[1];         // B*S = 16384
    const int blocks = (total_tokens + WAVES_PER_BLOCK - 1) / WAVES_PER_BLOCK;

    hipLaunchKernelGGL(NeuralPonds_54898271977921_kernel,
                       dim3(blocks), dim3(BLOCK_THREADS), 0, stream,
                       ctx, ponds, tables, out, total_tokens);
}